// GraphObservationEncoder_29214367547979
// MI455X (gfx1250) — compile-verified
//
#include <hip/hip_runtime.h>
#include <math.h>

// ---- problem constants (match reference) ----
#define HEADS 4
#define DIMH  64
#define HID   256           // HEADS*DIMH

typedef float v2f __attribute__((ext_vector_type(2)));
typedef float v8f __attribute__((ext_vector_type(8)));

// ---------------------------------------------------------------------------
// Generic f32 GEMM + bias (+optional ReLU) using V_WMMA_F32_16X16X4_F32.
// One wave computes one 16x16 output tile; K-loop steps by 4.
// Requires: M%16==0, N%16==0, K%4==0 (true for all call sites).
// A/B VGPR layout (ISA 7.12.2, 32-bit):  k = vreg + 2*(lane>>4),
//   A row = lane&15, B col = lane&15.  C/D: row = vreg + 8*(lane>>4).
// ---------------------------------------------------------------------------
__global__ __launch_bounds__(128)
void gemm_bias_wmma(const float* __restrict__ X, const float* __restrict__ W,
                    const float* __restrict__ bias, float* __restrict__ out,
                    int M, int K, int N, int out_stride, int do_relu)
{
    const int lane   = threadIdx.x & 31;
    const int wave   = threadIdx.x >> 5;
    const int tilesN = N >> 4;
    const int tiles  = (M >> 4) * tilesN;
    const int tile   = blockIdx.x * 4 + wave;     // wave-uniform
    if (tile >= tiles) return;                    // uniform exit -> EXEC all-1s at WMMA
    const int m0 = (tile / tilesN) << 4;
    const int n0 = (tile % tilesN) << 4;

    const int half = lane >> 4;                   // 0 or 1
    const int l15  = lane & 15;
    const int kgrp = half << 1;                   // 0 or 2

    v8f c = {0.f, 0.f, 0.f, 0.f, 0.f, 0.f, 0.f, 0.f};
    for (int k0 = 0; k0 < K; k0 += 4) {
        v2f a, b;
        const float* xrow = X + (size_t)(m0 + l15) * K + (k0 + kgrp);
        a.x = xrow[0];                            // k = k0+kgrp
        a.y = xrow[1];                            // k = k0+kgrp+1
        const float* wrow = W + (size_t)(k0 + kgrp) * N + (n0 + l15);
        b.x = wrow[0];
        b.y = wrow[N];
        c = __builtin_amdgcn_wmma_f32_16x16x4_f32(
                /*neg_a=*/false, a, /*neg_b=*/false, b,
                /*c_mod=*/(short)0, c, /*reuse_a=*/false, /*reuse_b=*/false);
    }

    const float bv = bias[n0 + l15];
#pragma unroll
    for (int r = 0; r < 8; ++r) {
        float v = c[r] + bv;
        if (do_relu) v = fmaxf(v, 0.f);
        out[(size_t)(m0 + r + half * 8) * out_stride + (n0 + l15)] = v;
    }
}

// ---------------------------------------------------------------------------
// init: segment-max buffers to -inf, denominators to 0
// ---------------------------------------------------------------------------
__global__ void init_softmax_bufs(float* __restrict__ mx_s, float* __restrict__ dn_s,
                                  float* __restrict__ mx_n, float* __restrict__ dn_n, int n)
{
    int i = blockIdx.x * blockDim.x + threadIdx.x;
    if (i < n) {
        mx_s[i] = -__builtin_inff();
        mx_n[i] = -__builtin_inff();
        dn_s[i] = 0.f;
        dn_n[i] = 0.f;
    }
}

__device__ inline void atomicMaxF(float* addr, float val)
{
    if (val >= 0.f) atomicMax((int*)addr, __float_as_int(val));
    else            atomicMin((unsigned int*)addr, __float_as_uint(val));
}

// ---------------------------------------------------------------------------
// Pass 1: per-edge attention logits + segment max.
// One wave per edge; lane covers 8 feature dims (2x float4), 8 lanes per head.
// ---------------------------------------------------------------------------
__global__ __launch_bounds__(256)
void edge_scores(const float* __restrict__ fs, const float* __restrict__ fd,
                 const float* __restrict__ attn,
                 const int* __restrict__ src, const int* __restrict__ dst,
                 float* __restrict__ scores, float* __restrict__ maxbuf, int E)
{
    const int lane = threadIdx.x & 31;
    const int e = blockIdx.x * (blockDim.x >> 5) + (threadIdx.x >> 5);
    if (e >= E) return;
    const int s  = src[e];
    const int dn = dst[e];
    const int base = lane * 8;                 // dims [base, base+7], one head per 8 lanes
    const int h    = lane >> 3;

    const float4* fsv = (const float4*)(fs + (size_t)s  * HID + base);
    const float4* fdv = (const float4*)(fd + (size_t)dn * HID + base);
    const float4* av  = (const float4*)(attn + h * DIMH + (base & (DIMH - 1)));

    float part = 0.f;
#pragma unroll
    for (int j = 0; j < 2; ++j) {
        float4 fa = fsv[j], fb = fdv[j], at = av[j];
        float v;
        v = fa.x + fb.x; v = (v >= 0.f) ? v : 0.2f * v; part += v * at.x;
        v = fa.y + fb.y; v = (v >= 0.f) ? v : 0.2f * v; part += v * at.y;
        v = fa.z + fb.z; v = (v >= 0.f) ? v : 0.2f * v; part += v * at.z;
        v = fa.w + fb.w; v = (v >= 0.f) ? v : 0.2f * v; part += v * at.w;
    }
    // reduce within each group of 8 lanes (one head)
    part += __shfl_xor(part, 1);
    part += __shfl_xor(part, 2);
    part += __shfl_xor(part, 4);
    if ((lane & 7) == 0) {
        scores[(size_t)e * HEADS + h] = part;
        atomicMaxF(&maxbuf[(size_t)dn * HEADS + h], part);
    }
}

// ---------------------------------------------------------------------------
// Pass 2: exponentiate (stable) + segment sum. One thread per (edge, head).
// ---------------------------------------------------------------------------
__global__ void edge_exp(const int* __restrict__ dst, float* __restrict__ scores,
                         const float* __restrict__ maxbuf, float* __restrict__ denom, int EH)
{
    int i = blockIdx.x * blockDim.x + threadIdx.x;
    if (i >= EH) return;
    const int e = i >> 2;
    const int h = i & 3;
    const int dn = dst[e];
    const float m  = maxbuf[dn * HEADS + h];   // finite: this dst has >=1 in-edge
    const float ex = __expf(scores[i] - m);
    scores[i] = ex;
    atomicAdd(&denom[dn * HEADS + h], ex);
}

// ---------------------------------------------------------------------------
// Pass 3: alpha-weighted scatter-add of fs[src] into xcat[dst, col_base:+256].
// One wave per edge; 8 f32 atomics per lane, coalesced across lanes.
// xcat rows were pre-seeded with the residual projection.
// ---------------------------------------------------------------------------
__global__ __launch_bounds__(256)
void edge_aggregate(const float* __restrict__ fs, const int* __restrict__ src,
                    const int* __restrict__ dst, const float* __restrict__ scores,
                    const float* __restrict__ denom, float* __restrict__ xcat,
                    int E, int col_base)
{
    const int lane = threadIdx.x & 31;
    const int e = blockIdx.x * (blockDim.x >> 5) + (threadIdx.x >> 5);
    if (e >= E) return;
    const int s  = src[e];
    const int dn = dst[e];

    float alpha[HEADS];
#pragma unroll
    for (int h = 0; h < HEADS; ++h)
        alpha[h] = scores[(size_t)e * HEADS + h] / denom[dn * HEADS + h];

    const float* fsrow = fs + (size_t)s * HID;
    float* orow = xcat + (size_t)dn * (2 * HID) + col_base;
#pragma unroll
    for (int i = 0; i < 8; ++i) {
        const int d = lane + i * 32;           // head = i/2 (compile-time per iter)
        atomicAdd(&orow[d], alpha[i >> 1] * fsrow[d]);
    }
}

__global__ void relu_inplace(float* __restrict__ x, int n)
{
    int i = blockIdx.x * blockDim.x + threadIdx.x;
    if (i < n) x[i] = fmaxf(x[i], 0.f);
}

// ---------------------------------------------------------------------------
extern "C" void kernel_launch(void* const* d_in, const int* in_sizes, int n_in,
                              void* d_out, int out_size, void* d_ws, size_t ws_size,
                              hipStream_t stream)
{
    const float* x_gt      = (const float*)d_in[0];
    const float* x_ubs     = (const float*)d_in[1];
    const float* x_ag      = (const float*)d_in[2];
    const int*   seen_src  = (const int*)d_in[3];
    const int*   seen_dst  = (const int*)d_in[4];
    const int*   near_src  = (const int*)d_in[5];
    const int*   near_dst  = (const int*)d_in[6];
    const float* Ws_seen   = (const float*)d_in[7];
    const float* bs_seen   = (const float*)d_in[8];
    const float* Wd_seen   = (const float*)d_in[9];
    const float* bd_seen   = (const float*)d_in[10];
    const float* attn_seen = (const float*)d_in[11];
    const float* Wr_seen   = (const float*)d_in[12];
    const float* br_seen   = (const float*)d_in[13];
    const float* Ws_near   = (const float*)d_in[14];
    const float* bs_near   = (const float*)d_in[15];
    const float* Wd_near   = (const float*)d_in[16];
    const float* bd_near   = (const float*)d_in[17];
    const float* attn_near = (const float*)d_in[18];
    const float* Wr_near   = (const float*)d_in[19];
    const float* br_near   = (const float*)d_in[20];
    const float* W_aggr    = (const float*)d_in[21];
    const float* b_aggr    = (const float*)d_in[22];

    const int N_GT = 100000, N_UBS = 20000, N_AG = 20000;
    const int E_SEEN = 320000, E_NEAR = 160000;
    const int F_GT = 8, F_UBS = 8, F_AG = 16;

    // workspace carve-up (all f32)
    float* ws = (float*)d_ws;
    float* fs_seen = ws; ws += (size_t)N_GT  * HID;      // 102.4 MB (L2-resident)
    float* fd_seen = ws; ws += (size_t)N_AG  * HID;
    float* fs_near = ws; ws += (size_t)N_UBS * HID;
    float* fd_near = ws; ws += (size_t)N_AG  * HID;
    float* xcat    = ws; ws += (size_t)N_AG  * 2 * HID;  // [N_AG, 512] concat buffer
    float* sc_seen = ws; ws += (size_t)E_SEEN * HEADS;
    float* sc_near = ws; ws += (size_t)E_NEAR * HEADS;
    float* mx_seen = ws; ws += (size_t)N_AG * HEADS;
    float* dn_seen = ws; ws += (size_t)N_AG * HEADS;
    float* mx_near = ws; ws += (size_t)N_AG * HEADS;
    float* dn_near = ws; ws += (size_t)N_AG * HEADS;

    // init softmax state
    {
        int n = N_AG * HEADS;
        init_softmax_bufs<<<(n + 255) / 256, 256, 0, stream>>>(mx_seen, dn_seen, mx_near, dn_near, n);
    }

    auto gemm = [&](const float* X, const float* W, const float* b, float* out,
                    int M, int K, int N, int stride, int relu) {
        int tiles = (M / 16) * (N / 16);
        gemm_bias_wmma<<<(tiles + 3) / 4, 128, 0, stream>>>(X, W, b, out, M, K, N, stride, relu);
    };

    // node-level linear projections (residuals land directly in xcat)
    gemm(x_gt,  Ws_seen, bs_seen, fs_seen,    N_GT,  F_GT,  HID, HID,     0);
    gemm(x_ag,  Wd_seen, bd_seen, fd_seen,    N_AG,  F_AG,  HID, HID,     0);
    gemm(x_ag,  Wr_seen, br_seen, xcat,       N_AG,  F_AG,  HID, 2 * HID, 0);
    gemm(x_ubs, Ws_near, bs_near, fs_near,    N_UBS, F_UBS, HID, HID,     0);
    gemm(x_ag,  Wd_near, bd_near, fd_near,    N_AG,  F_AG,  HID, HID,     0);
    gemm(x_ag,  Wr_near, br_near, xcat + HID, N_AG,  F_AG,  HID, 2 * HID, 0);

    // GATv2 edge softmax + aggregation, both relations
    edge_scores<<<(E_SEEN + 7) / 8, 256, 0, stream>>>(fs_seen, fd_seen, attn_seen,
                                                      seen_src, seen_dst, sc_seen, mx_seen, E_SEEN);
    edge_scores<<<(E_NEAR + 7) / 8, 256, 0, stream>>>(fs_near, fd_near, attn_near,
                                                      near_src, near_dst, sc_near, mx_near, E_NEAR);
    edge_exp<<<(E_SEEN * HEADS + 255) / 256, 256, 0, stream>>>(seen_dst, sc_seen, mx_seen, dn_seen, E_SEEN * HEADS);
    edge_exp<<<(E_NEAR * HEADS + 255) / 256, 256, 0, stream>>>(near_dst, sc_near, mx_near, dn_near, E_NEAR * HEADS);
    edge_aggregate<<<(E_SEEN + 7) / 8, 256, 0, stream>>>(fs_seen, seen_src, seen_dst,
                                                         sc_seen, dn_seen, xcat, E_SEEN, 0);
    edge_aggregate<<<(E_NEAR + 7) / 8, 256, 0, stream>>>(fs_near, near_src, near_dst,
                                                         sc_near, dn_near, xcat, E_NEAR, HID);

    relu_inplace<<<(N_AG * 2 * HID + 255) / 256, 256, 0, stream>>>(xcat, N_AG * 2 * HID);

    // fusion MLP: [20000,512] x [512,256] + bias, ReLU -> d_out
    gemm(xcat, W_aggr, b_aggr, (float*)d_out, N_AG, 2 * HID, HID, HID, 1);
}